// GNNWithVirtualNodeAndGINE_30116310679874
// MI455X (gfx1250) — compile-verified
//
#include <hip/hip_runtime.h>
#include <hip/hip_bf16.h>

#define Nn   50000
#define Ee   600000
#define Hh   128
#define EDd  16
#define Bb   256
#define Ll   5

typedef __attribute__((ext_vector_type(2))) float v2f;
typedef __attribute__((ext_vector_type(8))) float v8f;
typedef __attribute__((ext_vector_type(4))) int   v4i;

// ---------------------------------------------------------------------------
// Async global->LDS staging (gfx1250 GLOBAL_LOAD_ASYNC_TO_LDS_B128) with a
// synchronous fallback. 16 bytes per call.
// Builtin signature (from compiler diagnostic): arity 4, param0 = AS(1) v4i*.
// ---------------------------------------------------------------------------
#if __has_builtin(__builtin_amdgcn_global_load_async_to_lds_b128)
#define HAVE_ASYNC_LDS 1
typedef __attribute__((address_space(1))) v4i g4i;   // global
typedef __attribute__((address_space(3))) v4i l4i;   // LDS
#endif

__device__ __forceinline__ void cp16(float* dst_lds, const float* src_glb) {
#ifdef HAVE_ASYNC_LDS
    // int->ptr casts sidestep C++'s ban on cross-AS pointer casts.
    // LDS aperture lives in addr[63:32]; addr[31:0] is the LDS byte offset.
    g4i* gsrc = (g4i*)(unsigned long long)src_glb;
    l4i* ldst = (l4i*)(unsigned int)(unsigned long long)dst_lds;
    __builtin_amdgcn_global_load_async_to_lds_b128(gsrc, ldst, 0, 0);
#else
    *(float4*)dst_lds = *(const float4*)src_glb;
#endif
}

__device__ __forceinline__ void cp_wait() {
#ifdef HAVE_ASYNC_LDS
#if __has_builtin(__builtin_amdgcn_s_wait_asynccnt)
    __builtin_amdgcn_s_wait_asynccnt(0);
#else
    asm volatile("s_wait_asynccnt 0" ::: "memory");
#endif
#endif
}

// ---------------------------------------------------------------------------
// Generic fp32 WMMA GEMM: C[M,128] = op(A[M,128] @ W[128,128] + bias + extra)
//   extra (optional): extra[extra_idx[row]] row-broadcast add (vn[batch])
//   do_relu: relu AFTER bias/extra;  add_dst: C += result (vn residual)
// Grid: M/16 blocks of 256 threads (8 waves); wave w owns cols [16w,16w+16).
// M is always a multiple of 16 -> EXEC all-ones for WMMA.
// W is staged TRANSPOSED in LDS (Wt[n*34+k]) so both A and B operands are
// contiguous 8-byte ds loads (no VGPR repacking before v_wmma).
// ---------------------------------------------------------------------------
#define KP 34   // padded k-stride of transposed W chunk (even: keeps b64 align)

__global__ void wmma_gemm128(const float* __restrict__ A,
                             const float* __restrict__ W,
                             const float* __restrict__ bias,
                             const float* __restrict__ extra,
                             const int*   __restrict__ extra_idx,
                             float* __restrict__ C,
                             int do_relu, int add_dst)
{
    __shared__ float Alds[16 * 132];   // 16 rows x 128 K, padded stride
    __shared__ float Wt[128 * KP];     // 128 n x 32 k chunk, transposed

    const int tid = threadIdx.x;
    const int m0  = blockIdx.x * 16;

    // Stage this block's 16 A-rows fully (enables safe in-place C==A).
    for (int i = tid * 4; i < 16 * 128; i += 256 * 4) {
        const int r = i >> 7, c = i & 127;
        cp16(&Alds[r * 132 + c], &A[(size_t)(m0 + r) * 128 + c]);
    }
    cp_wait();

    const int wave = tid >> 5;
    const int lane = tid & 31;
    const int ln   = lane & 15;          // M (for A) / N (for B,C) within tile
    const int kh   = (lane >> 4) * 2;    // K sub-offset: 0 or 2
    const int n0   = wave * 16;

    v8f acc = {};

    for (int kb = 0; kb < 128; kb += 32) {
        __syncthreads();                  // protect Wt reuse (and A on kb==0)
        // Transpose-stage 32 K-rows of W: Wt[n*KP + kl] = W[kb+kl][n]
        for (int i = tid * 4; i < 32 * 128; i += 256 * 4) {
            const int kl = i >> 7, n = i & 127;
            const float4 w4 = *(const float4*)&W[(size_t)(kb + kl) * 128 + n];
            Wt[(n + 0) * KP + kl] = w4.x;
            Wt[(n + 1) * KP + kl] = w4.y;
            Wt[(n + 2) * KP + kl] = w4.z;
            Wt[(n + 3) * KP + kl] = w4.w;
        }
        __syncthreads();

        #pragma unroll
        for (int kk = 0; kk < 32; kk += 4) {
            const v2f a = *(const v2f*)&Alds[ln * 132 + kb + kk + kh];
            const v2f b = *(const v2f*)&Wt[(n0 + ln) * KP + kk + kh];
            acc = __builtin_amdgcn_wmma_f32_16x16x4_f32(
                false, a, false, b, (short)0, acc, false, false);
        }
    }

    // C layout: lane<16 -> row m0+v, col n0+ln ; lane>=16 -> row m0+8+v.
    const int rbase = m0 + ((lane >> 4) << 3);
    #pragma unroll
    for (int v = 0; v < 8; ++v) {
        const int row = rbase + v;
        float val = acc[v];
        if (bias)  val += bias[n0 + ln];
        if (extra) {
            const int er = extra_idx ? extra_idx[row] : row;
            val += extra[(size_t)er * 128 + n0 + ln];
        }
        if (do_relu) val = fmaxf(val, 0.0f);
        float* dst = &C[(size_t)row * 128 + n0 + ln];
        if (add_dst) val += *dst;
        *dst = val;
    }
}

// ---------------------------------------------------------------------------
// Fused edge pipeline: msg = h[src] + edge_attr @ eW + eb ; agg[dst] += msg
// One wave per edge: lane owns 4 contiguous features (float4). eW in LDS.
// ---------------------------------------------------------------------------
__global__ void edge_msg_scatter(const float* __restrict__ h,
                                 const float* __restrict__ edge_attr,
                                 const float* __restrict__ eW,
                                 const float* __restrict__ eb,
                                 const int*   __restrict__ src,
                                 const int*   __restrict__ dst,
                                 float* __restrict__ agg)
{
    __shared__ float eWs[EDd * 128];    // 8 KB
    const int tid = threadIdx.x;
    for (int i = tid * 4; i < EDd * 128; i += 256 * 4)
        cp16(&eWs[i], &eW[i]);
    cp_wait();
    __syncthreads();

    const int e = blockIdx.x * 8 + (tid >> 5);
    if (e >= Ee) return;
    const int lane = tid & 31;
    const int f = lane * 4;
    const int s = src[e], d = dst[e];

    float4 acc = *(const float4*)&h[(size_t)s * 128 + f];
    const float4 bv = *(const float4*)&eb[f];
    acc.x += bv.x; acc.y += bv.y; acc.z += bv.z; acc.w += bv.w;

    #pragma unroll
    for (int k = 0; k < EDd; ++k) {
        const float ea = edge_attr[(size_t)e * EDd + k];   // wave-broadcast
        const float4 w = *(const float4*)&eWs[k * 128 + f];
        acc.x += ea * w.x; acc.y += ea * w.y;
        acc.z += ea * w.z; acc.w += ea * w.w;
    }
    float* p = &agg[(size_t)d * 128 + f];
    unsafeAtomicAdd(p + 0, acc.x);
    unsafeAtomicAdd(p + 1, acc.y);
    unsafeAtomicAdd(p + 2, acc.z);
    unsafeAtomicAdd(p + 3, acc.w);
}

// ---------------------------------------------------------------------------
// Pooling / utility kernels
// ---------------------------------------------------------------------------
__global__ void zero_f32(float* p, long n) {
    long i = (long)blockIdx.x * blockDim.x + threadIdx.x;
    const long st = (long)gridDim.x * blockDim.x;
    for (; i < n; i += st) p[i] = 0.0f;
}

__global__ void zero_i32(int* p, int n) {
    int i = blockIdx.x * blockDim.x + threadIdx.x;
    if (i < n) p[i] = 0;
}

__global__ void bcast_row(float* vn, const float* __restrict__ row, int n) {
    int i = blockIdx.x * blockDim.x + threadIdx.x;
    if (i < n) vn[i] = row[i & 127];
}

__global__ void count_nodes(const int* __restrict__ batch, int* counts) {
    int i = blockIdx.x * blockDim.x + threadIdx.x;
    if (i < Nn) atomicAdd(&counts[batch[i]], 1);
}

__global__ void pool_sum(const float* __restrict__ x,
                         const int* __restrict__ batch,
                         float* __restrict__ pooled) {
    const int idx  = blockIdx.x * blockDim.x + threadIdx.x;
    const int node = idx >> 5;
    if (node >= Nn) return;
    const int f = (idx & 31) * 4;
    const float4 v = *(const float4*)&x[(size_t)node * 128 + f];
    float* p = &pooled[(size_t)batch[node] * 128 + f];
    unsafeAtomicAdd(p + 0, v.x);
    unsafeAtomicAdd(p + 1, v.y);
    unsafeAtomicAdd(p + 2, v.z);
    unsafeAtomicAdd(p + 3, v.w);
}

__global__ void pool_div(float* pooled, const int* __restrict__ counts, int n) {
    int i = blockIdx.x * blockDim.x + threadIdx.x;
    if (i < n) pooled[i] /= fmaxf((float)counts[i >> 7], 1.0f);
}

// ---------------------------------------------------------------------------
extern "C" void kernel_launch(void* const* d_in, const int* in_sizes, int n_in,
                              void* d_out, int out_size, void* d_ws, size_t ws_size,
                              hipStream_t stream) {
    const float* x         = (const float*)d_in[0];
    const float* edge_attr = (const float*)d_in[1];
    const float* node_W    = (const float*)d_in[2];
    const float* node_b    = (const float*)d_in[3];
    const float* edge_W    = (const float*)d_in[4];
    const float* edge_b    = (const float*)d_in[5];
    const float* mlp1_W    = (const float*)d_in[6];
    const float* mlp1_b    = (const float*)d_in[7];
    const float* mlp2_W    = (const float*)d_in[8];
    const float* mlp2_b    = (const float*)d_in[9];
    const float* vn_w0     = (const float*)d_in[10];
    const float* vn_b0     = (const float*)d_in[11];
    const float* vn_w1     = (const float*)d_in[12];
    const float* vn_b1     = (const float*)d_in[13];
    const float* fc_W      = (const float*)d_in[14];
    const float* fc_b      = (const float*)d_in[15];
    const float* vn_init   = (const float*)d_in[16];
    const int*   eidx      = (const int*)d_in[17];
    const int*   batch     = (const int*)d_in[18];
    const int*   src       = eidx;
    const int*   dstE      = eidx + Ee;
    float* out = (float*)d_out;

    char* ws = (char*)d_ws;
    float* buf_x   = (float*)ws;  ws += (size_t)Nn * Hh * 4;
    float* buf_agg = (float*)ws;  ws += (size_t)Nn * Hh * 4;
    float* vn      = (float*)ws;  ws += (size_t)Bb * Hh * 4;
    float* pooled  = (float*)ws;  ws += (size_t)Bb * Hh * 4;
    float* t1      = (float*)ws;  ws += (size_t)Bb * Hh * 4;
    int*   counts  = (int*)ws;    ws += (size_t)Bb * 4;

    // init (deterministic every call)
    zero_i32<<<1, 256, 0, stream>>>(counts, Bb);
    count_nodes<<<(Nn + 255) / 256, 256, 0, stream>>>(batch, counts);
    bcast_row<<<(Bb * Hh + 255) / 256, 256, 0, stream>>>(vn, vn_init, Bb * Hh);

    for (int l = 0; l < Ll; ++l) {
        const float* xin = (l == 0) ? x : buf_x;
        // h = x @ nW + nb + vn[batch]
        wmma_gemm128<<<Nn / 16, 256, 0, stream>>>(
            xin, node_W + (size_t)l * Hh * Hh, node_b + (size_t)l * Hh,
            vn, batch, buf_x, 0, 0);
        // agg = scatter_add(h[src] + edge_attr@eW + eb, dst)
        zero_f32<<<4096, 256, 0, stream>>>(buf_agg, (long)Nn * Hh);
        edge_msg_scatter<<<Ee / 8, 256, 0, stream>>>(
            buf_x, edge_attr, edge_W + (size_t)l * EDd * Hh,
            edge_b + (size_t)l * Hh, src, dstE, buf_agg);
        // t = relu(agg @ m1W + b)   (in place)
        wmma_gemm128<<<Nn / 16, 256, 0, stream>>>(
            buf_agg, mlp1_W + (size_t)l * Hh * Hh, mlp1_b + (size_t)l * Hh,
            nullptr, nullptr, buf_agg, 1, 0);
        // xn = relu(t @ m2W + b)
        wmma_gemm128<<<Nn / 16, 256, 0, stream>>>(
            buf_agg, mlp2_W + (size_t)l * Hh * Hh, mlp2_b + (size_t)l * Hh,
            nullptr, nullptr, buf_x, 1, 0);
        // pooled = segment_mean(xn, batch)
        zero_f32<<<64, 256, 0, stream>>>(pooled, (long)Bb * Hh);
        pool_sum<<<(Nn * 32 + 255) / 256, 256, 0, stream>>>(buf_x, batch, pooled);
        pool_div<<<(Bb * Hh + 255) / 256, 256, 0, stream>>>(pooled, counts, Bb * Hh);
        // vn += relu(relu(pooled @ w0 + b0) @ w1 + b1)
        wmma_gemm128<<<Bb / 16, 256, 0, stream>>>(
            pooled, vn_w0, vn_b0, nullptr, nullptr, t1, 1, 0);
        wmma_gemm128<<<Bb / 16, 256, 0, stream>>>(
            t1, vn_w1, vn_b1, nullptr, nullptr, vn, 1, 1);
    }

    // final mean pool + classifier
    zero_f32<<<64, 256, 0, stream>>>(pooled, (long)Bb * Hh);
    pool_sum<<<(Nn * 32 + 255) / 256, 256, 0, stream>>>(buf_x, batch, pooled);
    pool_div<<<(Bb * Hh + 255) / 256, 256, 0, stream>>>(pooled, counts, Bb * Hh);
    wmma_gemm128<<<Bb / 16, 256, 0, stream>>>(
        pooled, fc_W, fc_b, nullptr, nullptr, out, 0, 0);
}